// Learned_DMemOffsetBPDecoder_71545565216845
// MI455X (gfx1250) — compile-verified
//
#include <hip/hip_runtime.h>
#include <cstdint>

// Problem constants (match reference)
#define BB   128
#define CC   4096
#define VV   8192
#define DCC  8
#define DVV  4
#define EE   (CC * DCC)   // 32768
#define TT   10
#define BIGV 100000000.0f

#define NTHR 512

// ---------------------------------------------------------------------------
// CDNA5 async global<->LDS copies (ASYNCcnt path, gfx1250).
// Low 32 bits of a generic pointer to __shared__ data are the LDS byte
// address (aperture scheme) -- verified in round-1 asm
// (msg at LDS 0x0, prev at 0x20000).
// ---------------------------------------------------------------------------
__device__ __forceinline__ void async_copy_b128(void* lds_dst, const void* gsrc) {
    unsigned loff = (unsigned)(unsigned long long)(uintptr_t)lds_dst;
    asm volatile("global_load_async_to_lds_b128 %0, %1, off"
                 :: "v"(loff), "v"(gsrc)
                 : "memory");
}

__device__ __forceinline__ void async_store_b128(void* gdst, const void* lds_src) {
    unsigned loff = (unsigned)(unsigned long long)(uintptr_t)lds_src;
    asm volatile("global_store_async_from_lds_b128 %0, %1, off"
                 :: "v"(gdst), "v"(loff)
                 : "memory");
}

__device__ __forceinline__ void wait_asynccnt0() {
#if __has_builtin(__builtin_amdgcn_s_wait_asynccnt)
    __builtin_amdgcn_s_wait_asynccnt(0);
#else
    asm volatile("s_wait_asynccnt 0" ::: "memory");
#endif
}

// ---------------------------------------------------------------------------
// One workgroup per batch element. Entire per-batch BP state lives in LDS:
//   msg[E]  : v2c messages, overwritten in place by c2v each half-iteration
//   prev[V] : damped LLR memory == this iteration's output LLRs
// Graph structure: check i owns edges 8i..8i+7 (contiguous, ds_load_b128);
// variable j owns edges {j + m*V, m=0..3} (stride-V, conflict-free columns).
// Output out[b,t,:] is DMA'd from prev[] with async stores that overlap the
// next iteration's check phase.
// ---------------------------------------------------------------------------
__global__ __launch_bounds__(NTHR) void
bp_decoder_kernel(const int*   __restrict__ synd,    // (B, C) int32
                  const float* __restrict__ prior,   // (V)
                  const float* __restrict__ gamma_,  // (V)
                  const float* __restrict__ offset,  // (C, DC)
                  const float* __restrict__ nf,      // (C, DC)
                  float*       __restrict__ out)     // (B, T, V)
{
    __shared__ __align__(16) float msg[EE];   // 128 KB
    __shared__ __align__(16) float prev[VV];  //  32 KB  -> 160 KB total

    const int b   = blockIdx.x;
    const int tid = threadIdx.x;

    // ---- Init: broadcast prior into the 4 msg replicas + prev via async DMA
    for (int c4 = tid * 4; c4 < VV; c4 += NTHR * 4) {
        const float* g = prior + c4;
        async_copy_b128(&msg[c4],          g);
        async_copy_b128(&msg[VV + c4],     g);
        async_copy_b128(&msg[2 * VV + c4], g);
        async_copy_b128(&msg[3 * VV + c4], g);
        async_copy_b128(&prev[c4],         g);
    }
    wait_asynccnt0();
    __syncthreads();

    const int* sb = synd + (size_t)b * CC;

    for (int t = 0; t < TT; ++t) {
        // ================= Check-node update (min-sum, in place) ===========
        for (int i = tid; i < CC; i += NTHR) {
            const int e0 = i * DCC;
            float4 a0 = *(const float4*)&msg[e0];
            float4 a1 = *(const float4*)&msg[e0 + 4];
            float m[8] = {a0.x, a0.y, a0.z, a0.w, a1.x, a1.y, a1.z, a1.w};

            // exclusive sign product (exact zero handling) + exclusive min
            float pnz  = 1.0f;   // product of signs of nonzero messages
            int   zc   = 0;      // zero count
            float min1 = BIGV, min2 = BIGV;
            int   idx  = 0;      // first-occurrence argmin (matches jnp.argmin)
#pragma unroll
            for (int k = 0; k < 8; ++k) {
                float av = fabsf(m[k]);
                if (m[k] == 0.0f)      zc++;
                else if (m[k] < 0.0f)  pnz = -pnz;
                if (av < min1) { min2 = min1; min1 = av; idx = k; }
                else if (av < min2) { min2 = av; }
            }

            const float ssgn = 1.0f - 2.0f * (float)sb[i];
            float4 o0 = *(const float4*)(offset + e0);
            float4 o1 = *(const float4*)(offset + e0 + 4);
            float4 n0 = *(const float4*)(nf + e0);
            float4 n1 = *(const float4*)(nf + e0 + 4);
            float off8[8] = {o0.x, o0.y, o0.z, o0.w, o1.x, o1.y, o1.z, o1.w};
            float nf8[8]  = {n0.x, n0.y, n0.z, n0.w, n1.x, n1.y, n1.z, n1.w};

            float r0[8];
#pragma unroll
            for (int k = 0; k < 8; ++k) {
                float excl;
                if (zc == 0)      excl = (m[k] < 0.0f) ? -pnz : pnz; // P * s_k
                else if (zc == 1) excl = (m[k] == 0.0f) ? pnz : 0.0f;
                else              excl = 0.0f;
                float me = (k == idx) ? min2 : min1;      // exclusive min
                float r  = me - off8[k];
                r        = (r > 0.0f) ? r : 0.0f;         // relu
                r0[k]    = ssgn * excl * r * nf8[k];
            }
            float4 w0, w1;
            w0.x = r0[0]; w0.y = r0[1]; w0.z = r0[2]; w0.w = r0[3];
            w1.x = r0[4]; w1.y = r0[5]; w1.z = r0[6]; w1.w = r0[7];
            *(float4*)&msg[e0]     = w0;
            *(float4*)&msg[e0 + 4] = w1;
        }

        // Drain last iteration's async prev->out stores (and t=0 init loads)
        // in every wave, THEN barrier: after this point no async op can still
        // be reading prev[], so the var phase may overwrite it.
        wait_asynccnt0();
        __syncthreads();

        // ================= Variable-node update (in place) =================
        for (int j = tid; j < VV; j += NTHR) {
            float c0 = msg[j];
            float c1 = msg[VV + j];
            float c2 = msg[2 * VV + j];
            float c3 = msg[3 * VV + j];
            float g  = gamma_[j];
            float p  = prior[j];
            float pv = prev[j];
            float llr = (c0 + c1 + c2 + c3) + (1.0f - g) * p + g * pv;

            msg[j]          = llr - c0;   // extrinsic v2c
            msg[VV + j]     = llr - c1;
            msg[2 * VV + j] = llr - c2;
            msg[3 * VV + j] = llr - c3;
            prev[j]         = llr;        // == output LLR for step t
        }
        // Barrier commits all DS writes to prev[] (s_wait_dscnt 0 before
        // s_barrier) so the async engine reads finished data.
        __syncthreads();

        // ======= DMA prev[] -> out[b,t,:] ; overlaps next check phase ======
        {
            float* outb = out + (size_t)b * TT * VV + (size_t)t * VV;
            for (int c4 = tid * 4; c4 < VV; c4 += NTHR * 4) {
                async_store_b128(outb + c4, &prev[c4]);
            }
        }
        // No wait here: next check phase doesn't touch prev[]; the wait+
        // barrier at the top of the next var phase (or the implicit wait-idle
        // in S_ENDPGM on the last iteration) provides the drain.
    }
}

extern "C" void kernel_launch(void* const* d_in, const int* in_sizes, int n_in,
                              void* d_out, int out_size, void* d_ws, size_t ws_size,
                              hipStream_t stream) {
    const int*   synd   = (const int*)d_in[0];    // syndromes (B,C) int32
    const float* prior  = (const float*)d_in[1];  // prior_llr (V)
    const float* gamma_ = (const float*)d_in[2];  // gamma (V)
    const float* offset = (const float*)d_in[3];  // offset (C,DC)
    const float* nf     = (const float*)d_in[4];  // nf (C,DC)
    float*       out    = (float*)d_out;          // (B,T,V)

    hipLaunchKernelGGL(bp_decoder_kernel, dim3(BB), dim3(NTHR), 0, stream,
                       synd, prior, gamma_, offset, nf, out);
}